// GraphAttention_61469571940477
// MI455X (gfx1250) — compile-verified
//
#include <hip/hip_runtime.h>

#define T_DIM 48
#define S_DIM 128
#define N_NODES (T_DIM * S_DIM)   // 6144
#define IN_D 128
#define HEADS 4
#define HEAD_D 32
#define HD 128                    // HEADS*HEAD_D

typedef __attribute__((ext_vector_type(2))) float v2f;
typedef __attribute__((ext_vector_type(8))) float v8f;

__device__ __forceinline__ float lrelu02(float x) { return x > 0.f ? x : 0.2f * x; }

// ---------------------------------------------------------------------------
// Kernel 1: H = X @ W   (N x 128) = (N x 128)(128 x 128), fp32 WMMA 16x16x4
// block = 256 (8 waves): 16 rows x 128 cols per block; wave -> one 16x16 tile
// ---------------------------------------------------------------------------
__global__ void __launch_bounds__(256) gemm_xw(const float* __restrict__ X,
                                               const float* __restrict__ W,
                                               float* __restrict__ H) {
  const int lane  = threadIdx.x & 31;
  const int wv    = threadIdx.x >> 5;
  const int mbase = blockIdx.x * 16;
  const int nbase = wv * 16;
  const int mr    = lane & 15;          // A row / B,C col
  const int kh    = (lane >> 4) << 1;   // k pair: 0 or 2
  const float* arow = X + (mbase + mr) * IN_D;
  v8f c = {};
#pragma unroll 8
  for (int k = 0; k < IN_D; k += 4) {
    v2f a = *(const v2f*)(arow + k + kh);
    v2f b;
    b.x = W[(k + kh) * HD + nbase + mr];
    b.y = W[(k + kh + 1) * HD + nbase + mr];
    c = __builtin_amdgcn_wmma_f32_16x16x4_f32(false, a, false, b, (short)0, c,
                                              false, false);
  }
  const int mo = (lane >> 4) * 8;
#pragma unroll
  for (int i = 0; i < 8; ++i)
    H[(mbase + mo + i) * HD + nbase + mr] = c[i];
}

// ---------------------------------------------------------------------------
// Kernel 2: per-(node,head) attention coefficients a_src/a_dst = <h, att>
// ---------------------------------------------------------------------------
__global__ void attn_coeffs(const float* __restrict__ H,
                            const float* __restrict__ att_src,
                            const float* __restrict__ att_dst,
                            float* __restrict__ a_src,
                            float* __restrict__ a_dst) {
  int idx = blockIdx.x * blockDim.x + threadIdx.x;  // node*HEADS + h
  if (idx >= N_NODES * HEADS) return;
  int node = idx >> 2;
  int h = idx & 3;
  const float* hp = H + node * HD + h * HEAD_D;
  const float* as = att_src + h * HEAD_D;
  const float* ad = att_dst + h * HEAD_D;
  float s = 0.f, d = 0.f;
#pragma unroll
  for (int i = 0; i < HEAD_D; ++i) { float v = hp[i]; s += v * as[i]; d += v * ad[i]; }
  a_src[idx] = s;
  a_dst[idx] = d;
}

// ---------------------------------------------------------------------------
// Kernel 3: per-(dst,head) softmax max and denominator over 175 neighbors
// ---------------------------------------------------------------------------
__global__ void softmax_stats(const float* __restrict__ a_src,
                              const float* __restrict__ a_dst,
                              float* __restrict__ mmax,
                              float* __restrict__ denom) {
  int idx = blockIdx.x * blockDim.x + threadIdx.x;
  if (idx >= N_NODES * HEADS) return;
  int node = idx >> 2;
  int h = idx & 3;
  int r = node / S_DIM, c = node % S_DIM;
  float ad = a_dst[idx];
  float m = -3.4e38f;
  for (int c2 = 0; c2 < S_DIM; ++c2)
    m = fmaxf(m, lrelu02(a_src[(r * S_DIM + c2) * HEADS + h] + ad));
  for (int t = 0; t < T_DIM; ++t) {
    if (t == r) continue;
    m = fmaxf(m, lrelu02(a_src[(t * S_DIM + c) * HEADS + h] + ad));
  }
  float den = 0.f;
  for (int c2 = 0; c2 < S_DIM; ++c2)
    den += __expf(lrelu02(a_src[(r * S_DIM + c2) * HEADS + h] + ad) - m);
  for (int t = 0; t < T_DIM; ++t) {
    if (t == r) continue;
    den += __expf(lrelu02(a_src[(t * S_DIM + c) * HEADS + h] + ad) - m);
  }
  mmax[idx] = m;
  denom[idx] = den;
}

// ---------------------------------------------------------------------------
// Kernel 4: row aggregation via WMMA. Per (r,h): EX[128x128] x Hrow[128x32].
// block = 512 (16 waves): wave -> (mtile 0..7, ntile 0..1), A built on the fly
// ---------------------------------------------------------------------------
__global__ void __launch_bounds__(512) row_aggregate(const float* __restrict__ H,
                                                     const float* __restrict__ a_src,
                                                     const float* __restrict__ a_dst,
                                                     const float* __restrict__ mmax,
                                                     float* __restrict__ OUTR) {
  const int r = blockIdx.x;          // 0..47
  const int h = blockIdx.y;          // 0..3
  const int wv = threadIdx.x >> 5;   // 0..15
  const int lane = threadIdx.x & 31;
  const int mtile = wv & 7;
  const int ntile = wv >> 3;
  const int m = lane & 15;
  const int kh = (lane >> 4) << 1;
  const int dst = r * S_DIM + mtile * 16 + m;
  const float ad = a_dst[dst * HEADS + h];
  const float mm = mmax[dst * HEADS + h];
  const int nb = h * HEAD_D + ntile * 16 + m;   // B/C column offset in H row
  v8f c = {};
#pragma unroll 4
  for (int k = 0; k < S_DIM; k += 4) {
    int s0 = r * S_DIM + k + kh;
    v2f a, b;
    a.x = __expf(lrelu02(a_src[s0 * HEADS + h] + ad) - mm);
    a.y = __expf(lrelu02(a_src[(s0 + 1) * HEADS + h] + ad) - mm);
    b.x = H[s0 * HD + nb];
    b.y = H[(s0 + 1) * HD + nb];
    c = __builtin_amdgcn_wmma_f32_16x16x4_f32(false, a, false, b, (short)0, c,
                                              false, false);
  }
  const int mo = (lane >> 4) * 8;
#pragma unroll
  for (int i = 0; i < 8; ++i)
    OUTR[(r * S_DIM + mtile * 16 + mo + i) * HD + nb] = c[i];
}

// ---------------------------------------------------------------------------
// Kernel 5: column aggregation via WMMA. Per (c,h): EX[48x48] x Hcol[48x32],
// diagonal (src row == dst row) masked to 0. block = 192 (6 waves).
// ---------------------------------------------------------------------------
__global__ void __launch_bounds__(192) col_aggregate(const float* __restrict__ H,
                                                     const float* __restrict__ a_src,
                                                     const float* __restrict__ a_dst,
                                                     const float* __restrict__ mmax,
                                                     float* __restrict__ OUTC) {
  const int cidx = blockIdx.x;       // 0..127
  const int h = blockIdx.y;          // 0..3
  const int wv = threadIdx.x >> 5;   // 0..5
  const int lane = threadIdx.x & 31;
  const int mtile = wv % 3;          // 48 dst rows = 3 tiles of 16
  const int ntile = wv / 3;
  const int m = lane & 15;
  const int kh = (lane >> 4) << 1;
  const int rdst = mtile * 16 + m;
  const int dst = rdst * S_DIM + cidx;
  const float ad = a_dst[dst * HEADS + h];
  const float mm = mmax[dst * HEADS + h];
  const int nb = h * HEAD_D + ntile * 16 + m;
  v8f c = {};
#pragma unroll 3
  for (int k = 0; k < T_DIM; k += 4) {
    int t0 = k + kh;
    int n0 = t0 * S_DIM + cidx;
    int n1 = n0 + S_DIM;
    v2f a, b;
    float e0 = __expf(lrelu02(a_src[n0 * HEADS + h] + ad) - mm);
    float e1 = __expf(lrelu02(a_src[n1 * HEADS + h] + ad) - mm);
    a.x = (t0 == rdst) ? 0.f : e0;
    a.y = (t0 + 1 == rdst) ? 0.f : e1;
    b.x = H[n0 * HD + nb];
    b.y = H[n1 * HD + nb];
    c = __builtin_amdgcn_wmma_f32_16x16x4_f32(false, a, false, b, (short)0, c,
                                              false, false);
  }
  const int mo = (lane >> 4) * 8;
#pragma unroll
  for (int i = 0; i < 8; ++i)
    OUTC[((mtile * 16 + mo + i) * S_DIM + cidx) * HD + nb] = c[i];
}

// ---------------------------------------------------------------------------
// Kernel 6: Y = ELU((OUTR + OUTC)/denom + bias)
// ---------------------------------------------------------------------------
__global__ void combine_elu(const float* __restrict__ OUTR,
                            const float* __restrict__ OUTC,
                            const float* __restrict__ denom,
                            const float* __restrict__ bias,
                            float* __restrict__ Y) {
  int e = blockIdx.x * blockDim.x + threadIdx.x;
  if (e >= N_NODES * HD) return;
  int node = e >> 7;
  int f = e & 127;
  int h = f >> 5;
  float v = (OUTR[e] + OUTC[e]) / denom[node * HEADS + h] + bias[f];
  Y[e] = v > 0.f ? v : __expf(v) - 1.f;
}

// ---------------------------------------------------------------------------
extern "C" void kernel_launch(void* const* d_in, const int* in_sizes, int n_in,
                              void* d_out, int out_size, void* d_ws, size_t ws_size,
                              hipStream_t stream) {
  const float* X0  = (const float*)d_in[0];   // [1,48,128,128] == [N,128]
  const float* W1  = (const float*)d_in[1];
  const float* as1 = (const float*)d_in[2];
  const float* ad1 = (const float*)d_in[3];
  const float* b1  = (const float*)d_in[4];
  const float* W2  = (const float*)d_in[5];
  const float* as2 = (const float*)d_in[6];
  const float* ad2 = (const float*)d_in[7];
  const float* b2  = (const float*)d_in[8];
  float* out = (float*)d_out;

  float* ws = (float*)d_ws;
  const int NF = N_NODES * HD;      // 786432
  const int NH = N_NODES * HEADS;   // 24576
  float* H    = ws;
  float* OUTR = ws + (size_t)NF;
  float* OUTC = ws + (size_t)2 * NF;
  float* X1   = ws + (size_t)3 * NF;
  float* a_s  = ws + (size_t)4 * NF;
  float* a_d  = a_s + NH;
  float* mm   = a_d + NH;
  float* den  = mm + NH;

  // ---- layer 1
  gemm_xw<<<N_NODES / 16, 256, 0, stream>>>(X0, W1, H);
  attn_coeffs<<<(NH + 255) / 256, 256, 0, stream>>>(H, as1, ad1, a_s, a_d);
  softmax_stats<<<(NH + 255) / 256, 256, 0, stream>>>(a_s, a_d, mm, den);
  row_aggregate<<<dim3(T_DIM, HEADS), 512, 0, stream>>>(H, a_s, a_d, mm, OUTR);
  col_aggregate<<<dim3(S_DIM, HEADS), 192, 0, stream>>>(H, a_s, a_d, mm, OUTC);
  combine_elu<<<(NF + 255) / 256, 256, 0, stream>>>(OUTR, OUTC, den, b1, X1);

  // ---- layer 2
  gemm_xw<<<N_NODES / 16, 256, 0, stream>>>(X1, W2, H);
  attn_coeffs<<<(NH + 255) / 256, 256, 0, stream>>>(H, as2, ad2, a_s, a_d);
  softmax_stats<<<(NH + 255) / 256, 256, 0, stream>>>(a_s, a_d, mm, den);
  row_aggregate<<<dim3(T_DIM, HEADS), 512, 0, stream>>>(H, a_s, a_d, mm, OUTR);
  col_aggregate<<<dim3(S_DIM, HEADS), 192, 0, stream>>>(H, a_s, a_d, mm, OUTC);
  combine_elu<<<(NF + 255) / 256, 256, 0, stream>>>(OUTR, OUTC, den, b2, out);
}